// S4DKernel_16363825398526
// MI455X (gfx1250) — compile-verified
//
#include <hip/hip_runtime.h>
#include <math.h>

// S4D convolution-kernel generation: K[d,l] = Re( sum_n w[d,n] * p[d,n]^l )
// Restructured as two real 64x64x64 GEMMs per channel via l = m*64 + j,
// executed with v_wmma_f32_16x16x4_f32 (fp32 tensor path for complex64 accuracy).
//
// Transcendental budget: all phases are dAi * (integer t); residues mod 2*pi
// are pre-reduced in double once per state so the hot loop only needs the
// native v_sin/v_cos/v_exp (|ph| < 2*pi*63 ~ 396 rad < 256*pi native range).
// B-matrix N-tiles are related by B(j+16) = B(j) * p^16, so tiles 1..3 are
// produced by chained complex multiplies (full-rate VALU) instead of
// exp+sincos (quarter-rate TRANS): 12 TRANS per k-step instead of 30.

#define D_MODEL 1024
#define D_STATE 64
#define SEQ_L   4096

typedef __attribute__((ext_vector_type(2))) float v2f;
typedef __attribute__((ext_vector_type(8))) float v8f;

__global__ __launch_bounds__(128) void s4d_wmma_kernel(
    const float* __restrict__ log_dt,      // (D)
    const float* __restrict__ log_A_real,  // (D,N)
    const float* __restrict__ A_imag,      // (D,N)
    const float* __restrict__ Bc,          // (D,N) complex64 interleaved
    const float* __restrict__ Cc,          // (D,N) complex64 interleaved
    float* __restrict__ out)               // (D,L) float32
{
    __shared__ float sh_dtAr[D_STATE];   // dt * Re(A)
    __shared__ float sh_r1[D_STATE];     // (dt * Im(A)) mod 2pi
    __shared__ float sh_r64[D_STATE];    // (64 * dt * Im(A)) mod 2pi
    __shared__ float sh_wr[D_STATE];     // Re(C * B_bar)
    __shared__ float sh_wi[D_STATE];     // Im(C * B_bar)
    __shared__ float sh_p16r[D_STATE];   // Re(p^16)
    __shared__ float sh_p16i[D_STATE];   // Im(p^16)

    const int d   = blockIdx.x;
    const int tid = threadIdx.x;

    // ---- Phase 1: per-state discretization: dtA, w = C*B_bar, residues, p^16 ----
    if (tid < D_STATE) {
        const int n   = tid;
        const int idx = d * D_STATE + n;
        const float dt  = __expf(log_dt[d]);
        const float ar  = -__expf(log_A_real[idx]);
        const float ai  = A_imag[idx];
        const float dtar = ar * dt;
        const float dtai = ai * dt;
        // p = exp(dtA)   (|dtai| <= ~20 rad: native sincos is fine)
        const float er = __expf(dtar);
        float s, c;
        __sincosf(dtai, &s, &c);
        const float pr = er * c, pi = er * s;
        // B_bar = (p - 1)/A * B   (divide via conj(A)/|A|^2)
        const float inv  = 1.0f / (ar * ar + ai * ai);
        const float pm1r = pr - 1.0f;
        const float qr = (pm1r * ar + pi * ai) * inv;
        const float qi = (pi * ar - pm1r * ai) * inv;
        const float br = Bc[2 * idx], bi = Bc[2 * idx + 1];
        const float bbr = qr * br - qi * bi;
        const float bbi = qr * bi + qi * br;
        // w = C * B_bar
        const float cr = Cc[2 * idx], ci = Cc[2 * idx + 1];
        sh_wr[n] = cr * bbr - ci * bbi;
        sh_wi[n] = cr * bbi + ci * bbr;
        sh_dtAr[n] = dtar;
        // Phase residues mod 2*pi (double fmod: exact enough; once per state).
        const double TWO_PI = 6.2831853071795864769252867665590;
        sh_r1[n]  = (float)fmod((double)dtai, TWO_PI);
        sh_r64[n] = (float)fmod(64.0 * (double)dtai, TWO_PI);
        // p^16 = exp(16*dtA): ratio between adjacent B N-tiles.
        const float e16 = __expf(16.0f * dtar);
        const float r16 = (float)fmod(16.0 * (double)dtai, TWO_PI);
        float s16, c16;
        __sincosf(r16, &s16, &c16);
        sh_p16r[n] = e16 * c16;
        sh_p16i[n] = e16 * s16;
    }
    __syncthreads();

    // ---- Phase 2: per-wave 16x64 output strip via WMMA over K = 64 states ----
    const int wave = tid >> 5;      // M-tile index (rows m = 16*wave .. +15)
    const int lane = tid & 31;
    const int half = lane >> 4;     // 0: K slots {0,1};  1: K slots {2,3}
    const int fr   = lane & 15;     // free index (M for A, N for B)

    v8f acc0 = {}, acc1 = {}, acc2 = {}, acc3 = {};

    const float mf  = (float)(wave * 16 + fr);   // A row index m in [0,64)
    const float m64 = mf * 64.0f;                // exponent multiplier for A
    const float jf  = (float)fr;                 // B column within N-tile 0

    for (int kk = 0; kk < 16; ++kk) {
        const int kb = kk * 4 + half * 2;  // this lane's two K (state) slots
        const float dAr0 = sh_dtAr[kb],   dAr1 = sh_dtAr[kb + 1];
        const float r1_0 = sh_r1[kb],     r1_1 = sh_r1[kb + 1];
        const float r64_0 = sh_r64[kb],   r64_1 = sh_r64[kb + 1];
        const float wr0  = sh_wr[kb],     wi0  = sh_wi[kb];
        const float wr1  = sh_wr[kb + 1], wi1  = sh_wi[kb + 1];
        const float g0r  = sh_p16r[kb],   g0i  = sh_p16i[kb];
        const float g1r  = sh_p16r[kb+1], g1i  = sh_p16i[kb+1];

        // A operand: w * p^(64m); real part in Ar, NEGATED imag part in An
        // (so acc += Ar*Br + An*Bi realizes Re = Ur*Wr - Ui*Wi).
        v2f Ar, An;
        {
            float er = __expf(dAr0 * m64);
            float s, c;
            __sincosf(r64_0 * mf, &s, &c);   // |ph| < 396 rad, native-safe
            float vr = er * c, vi = er * s;
            Ar.x = wr0 * vr - wi0 * vi;
            An.x = -(wr0 * vi + wi0 * vr);
            er = __expf(dAr1 * m64);
            __sincosf(r64_1 * mf, &s, &c);
            vr = er * c; vi = er * s;
            Ar.y = wr1 * vr - wi1 * vi;
            An.y = -(wr1 * vi + wi1 * vr);
        }

        // B N-tile 0: p^j via exp/sincos (only TRANS use for B).
        v2f Br, Bi;
        {
            float er = __expf(dAr0 * jf);
            float s, c;
            __sincosf(r1_0 * jf, &s, &c);
            Br.x = er * c; Bi.x = er * s;
            er = __expf(dAr1 * jf);
            __sincosf(r1_1 * jf, &s, &c);
            Br.y = er * c; Bi.y = er * s;
        }
        acc0 = __builtin_amdgcn_wmma_f32_16x16x4_f32(
            false, Ar, false, Br, (short)0, acc0, false, false);
        acc0 = __builtin_amdgcn_wmma_f32_16x16x4_f32(
            false, An, false, Bi, (short)0, acc0, false, false);

        // B N-tiles 1..3: chained complex multiply by p^16 (full-rate VALU).
#define S4D_STEP(ACC)                                                          \
        {                                                                      \
            const float t0r = Br.x * g0r - Bi.x * g0i;                         \
            const float t0i = Br.x * g0i + Bi.x * g0r;                         \
            const float t1r = Br.y * g1r - Bi.y * g1i;                         \
            const float t1i = Br.y * g1i + Bi.y * g1r;                         \
            Br.x = t0r; Bi.x = t0i; Br.y = t1r; Bi.y = t1i;                    \
            ACC = __builtin_amdgcn_wmma_f32_16x16x4_f32(                       \
                false, Ar, false, Br, (short)0, ACC, false, false);            \
            ACC = __builtin_amdgcn_wmma_f32_16x16x4_f32(                       \
                false, An, false, Bi, (short)0, ACC, false, false);            \
        }
        S4D_STEP(acc1)
        S4D_STEP(acc2)
        S4D_STEP(acc3)
#undef S4D_STEP
    }

    // ---- Store: C/D layout -> row = r + 8*half, col = fr (per 16x16 tile) ----
    float* __restrict__ o = out + (size_t)d * SEQ_L;
#pragma unroll
    for (int r = 0; r < 8; ++r) {
        const int row = wave * 16 + half * 8 + r;   // m in [0,64)
        float* __restrict__ orow = o + row * 64;    // l = m*64 + j
        orow[ 0 + fr] = acc0[r];
        orow[16 + fr] = acc1[r];
        orow[32 + fr] = acc2[r];
        orow[48 + fr] = acc3[r];
    }
}

extern "C" void kernel_launch(void* const* d_in, const int* in_sizes, int n_in,
                              void* d_out, int out_size, void* d_ws, size_t ws_size,
                              hipStream_t stream) {
    (void)in_sizes; (void)n_in; (void)d_ws; (void)ws_size; (void)out_size;
    const float* log_dt     = (const float*)d_in[0];
    const float* log_A_real = (const float*)d_in[1];
    const float* A_imag     = (const float*)d_in[2];
    const float* Bc         = (const float*)d_in[3];  // complex64 -> interleaved f32
    const float* Cc         = (const float*)d_in[4];  // complex64 -> interleaved f32
    float* out              = (float*)d_out;

    dim3 grid(D_MODEL);
    dim3 block(128);  // 4 wave32 waves: one 16-row M strip each
    s4d_wmma_kernel<<<grid, block, 0, stream>>>(log_dt, log_A_real, A_imag, Bc, Cc, out);
}